// PanopticHead_39633958207557
// MI455X (gfx1250) — compile-verified
//
#include <hip/hip_runtime.h>

// Problem constants (from reference)
#define Nn   128
#define Hh   400
#define Wd   400
#define MSz  28
#define CTn  80
#define CSn  53
#define HWn  (Hh*Wd)          // 160000
#define NCH  (CSn+Nn)         // 181
#define WGn  1600
#define IGNORE_L 255

typedef __attribute__((ext_vector_type(2))) float v2f;
typedef __attribute__((ext_vector_type(8))) float v8f;

// ---------------------------------------------------------------------------
// Kernel 1: copy stuff_logit (53x400x400) into channels [0..52] of pan_logit.
// Pure streaming copy, float4 vectorized (b128 loads/stores).
// ---------------------------------------------------------------------------
__global__ __launch_bounds__(256) void stuff_copy_kernel(const float* __restrict__ stuff,
                                                         float* __restrict__ out) {
  size_t i = (size_t)blockIdx.x * blockDim.x + threadIdx.x;
  size_t total4 = (size_t)CSn * HWn / 4;   // 2,120,000 float4s
  if (i < total4) {
    ((float4*)out)[i] = ((const float4*)stuff)[i];
  }
}

// ---------------------------------------------------------------------------
// Kernel 2: per-instance thing channel.  val = Wy * ml * Wx^T computed as
// 16x16 output tiles via V_WMMA_F32_16X16X4_F32 (K=28 -> 7 k-steps of 4).
// Then box-mask, add thing semantic logit in its own box, store full plane.
// grid = (79, 128), block = 256 (8 waves, one 16x16 tile per wave).
// ---------------------------------------------------------------------------
__global__ __launch_bounds__(256) void things_kernel(const float* __restrict__ mask_logits,
                                                     const float* __restrict__ thing_logit,
                                                     const float* __restrict__ bbox,
                                                     const int*   __restrict__ cls_idx,
                                                     float* __restrict__ out) {
  __shared__ float s_ml[MSz * MSz];

  const int n   = blockIdx.y;
  const int cls = cls_idx[n];

  // stage the selected 28x28 mask tile into LDS
  for (int i = threadIdx.x; i < MSz * MSz; i += 256)
    s_ml[i] = mask_logits[((size_t)n * CTn + cls) * (MSz * MSz) + i];
  __syncthreads();

  const int wave = threadIdx.x >> 5;
  const int lane = threadIdx.x & 31;
  const int tile = blockIdx.x * 8 + wave;          // uniform per wave
  if (tile >= 25 * 25) return;                     // wave-uniform exit (EXEC stays all-1)

  const int ty0 = (tile / 25) * 16;
  const int tx0 = (tile % 25) * 16;

  // per-instance scalars (cheap, recomputed per thread)
  const float bx0 = bbox[n * 4 + 0], by0 = bbox[n * 4 + 1];
  const float bx2 = bbox[n * 4 + 2], by2 = bbox[n * 4 + 3];
  const int x0b = (int)floorf(bx0 * 0.25f), y0b = (int)floorf(by0 * 0.25f);
  const int x2b = (int)floorf(bx2 * 0.25f), y2b = (int)floorf(by2 * 0.25f);
  const int hh = y2b - y0b + 1, ww = x2b - x0b + 1;
  const int cx1 = (int)(bx0 * 0.25f), cy1 = (int)(by0 * 0.25f);          // trunc == floor (>=0)
  const int cx2 = (int)(rintf(bx2 * 0.25f) + 1.0f);                      // jnp.round = rint (RNE)
  const int cy2 = (int)(rintf(by2 * 0.25f) + 1.0f);

  const int m16  = lane & 15;
  const int koff = (lane < 16) ? 0 : 2;   // A/B K-split across half-waves per ISA layout

  // y-axis sample params for this lane's output row (A operand source)
  const int   yl   = ty0 + m16;
  const float srcy = ((float)(yl - y0b) + 0.5f) * ((float)MSz / (float)hh) - 0.5f;
  const float fy   = floorf(srcy);
  const float tyf  = srcy - fy;
  const int   i0   = (int)fminf(fmaxf(fy,        0.0f), (float)(MSz - 1));
  const int   i1   = (int)fminf(fmaxf(fy + 1.0f, 0.0f), (float)(MSz - 1));
  const float wy0  = 1.0f - tyf, wy1 = tyf;

  // x-axis sample params for this lane's output column (B operand source)
  const int   xl   = tx0 + m16;
  const float srcx = ((float)(xl - x0b) + 0.5f) * ((float)MSz / (float)ww) - 0.5f;
  const float fx   = floorf(srcx);
  const float txf  = srcx - fx;
  const int   j0   = (int)fminf(fmaxf(fx,        0.0f), (float)(MSz - 1));
  const int   j1   = (int)fminf(fmaxf(fx + 1.0f, 0.0f), (float)(MSz - 1));

  v8f acc = {};   // C/D: 16x16 f32 accumulator
#pragma unroll
  for (int s = 0; s < 7; ++s) {
    const int k0 = 4 * s + koff;
    v2f a, b;
    // A(16x4): lane m holds rows[ty0+m, k0] / rows[ty0+m, k0+1]
    a.x = s_ml[i0 * MSz + k0    ] * wy0 + s_ml[i1 * MSz + k0    ] * wy1;
    a.y = s_ml[i0 * MSz + k0 + 1] * wy0 + s_ml[i1 * MSz + k0 + 1] * wy1;
    // B(4x16): sparse x-interp weights for column x = tx0+m16
    b.x = ((k0     == j0) ? (1.0f - txf) : 0.0f) + ((k0     == j1) ? txf : 0.0f);
    b.y = ((k0 + 1 == j0) ? (1.0f - txf) : 0.0f) + ((k0 + 1 == j1) ? txf : 0.0f);
    acc = __builtin_amdgcn_wmma_f32_16x16x4_f32(false, a, false, b, (short)0, acc,
                                                false, false);
  }

  // D layout: VGPR r -> (M=r, lanes 0-15) / (M=r+8, lanes 16-31); N = lane&15
  const int vy_lo = max(y0b, 0), vy_hi = min(y2b, Hh - 1);
  const int vx_lo = max(x0b, 0), vx_hi = min(x2b, Wd - 1);
  const int yadd  = (lane >= 16) ? 8 : 0;
#pragma unroll
  for (int r = 0; r < 8; ++r) {
    const int y = ty0 + r + yadd;
    const int x = tx0 + m16;
    float o = (y >= vy_lo && y <= vy_hi && x >= vx_lo && x <= vx_hi) ? acc[r] : 0.0f;
    if (y >= cy1 && y < cy2 && x >= cx1 && x < cx2)
      o += thing_logit[(size_t)cls * HWn + (size_t)y * Wd + x];
    out[(size_t)(CSn + n) * HWn + (size_t)y * Wd + x] = o;
  }
}

// ---------------------------------------------------------------------------
// Kernel 3: per-pixel log-softmax over 181 channels + masked NLL accumulation.
// Two channel passes; pan_logit (116 MB) fits in the 192 MB L2, so pass 2 hits
// cache.  Block tree-reduction then 2 float atomics per block into workspace.
// ---------------------------------------------------------------------------
__global__ __launch_bounds__(256) void loss_kernel(const float* __restrict__ out,
                                                   const int*   __restrict__ gt,
                                                   float* __restrict__ accum) {
  const int p = blockIdx.x * 256 + threadIdx.x;
  float nll = 0.0f, cnt = 0.0f;
  if (p < HWn) {
    const int y = p / Wd, x = p - y * Wd;
    const int g = gt[(size_t)(4 * y) * WGn + 4 * x];      // exact nearest downsample
    float m = -3.4028235e38f;
    for (int c = 0; c < NCH; ++c) m = fmaxf(m, out[(size_t)c * HWn + p]);
    float ssum = 0.0f;
    for (int c = 0; c < NCH; ++c) ssum += __expf(out[(size_t)c * HWn + p] - m);
    const int tgt = min(max(g, 0), NCH - 1);
    const float lse = m + __logf(ssum);
    const float vt  = out[(size_t)tgt * HWn + p];
    if (g != IGNORE_L) { nll = lse - vt; cnt = 1.0f; }
  }
  __shared__ float s1[256], s2[256];
  s1[threadIdx.x] = nll; s2[threadIdx.x] = cnt;
  __syncthreads();
  for (int st = 128; st > 0; st >>= 1) {
    if (threadIdx.x < st) {
      s1[threadIdx.x] += s1[threadIdx.x + st];
      s2[threadIdx.x] += s2[threadIdx.x + st];
    }
    __syncthreads();
  }
  if (threadIdx.x == 0) {
    atomicAdd(&accum[0], s1[0]);
    atomicAdd(&accum[1], s2[0]);
  }
}

__global__ void final_kernel(const float* __restrict__ accum, float* __restrict__ out) {
  out[(size_t)NCH * HWn] = 1.0f /*LOSS_W*/ * accum[0] / fmaxf(accum[1], 1.0f);
}

// ---------------------------------------------------------------------------
extern "C" void kernel_launch(void* const* d_in, const int* in_sizes, int n_in,
                              void* d_out, int out_size, void* d_ws, size_t ws_size,
                              hipStream_t stream) {
  const float* mask_logits = (const float*)d_in[0];   // (128,80,28,28) f32
  const float* stuff_logit = (const float*)d_in[1];   // (53,400,400)  f32
  const float* thing_logit = (const float*)d_in[2];   // (80,400,400)  f32
  const float* bbox        = (const float*)d_in[3];   // (128,4)       f32
  const int*   cls_idx     = (const int*)d_in[4];     // (128,)        i32
  const int*   gt          = (const int*)d_in[5];     // (1600,1600)   i32
  float* out   = (float*)d_out;                       // 181*160000 + 1 f32
  float* accum = (float*)d_ws;

  hipMemsetAsync(d_ws, 0, 2 * sizeof(float), stream);  // graph-capturable

  stuff_copy_kernel<<<((CSn * HWn / 4) + 255) / 256, 256, 0, stream>>>(stuff_logit, out);

  dim3 g((625 + 7) / 8, Nn);   // 79 x 128 blocks, 8 waves/block, 1 tile/wave
  things_kernel<<<g, 256, 0, stream>>>(mask_logits, thing_logit, bbox, cls_idx, out);

  loss_kernel<<<(HWn + 255) / 256, 256, 0, stream>>>(out, gt, accum);
  final_kernel<<<1, 1, 0, stream>>>(accum, out);
}